// TRNNCell_28535762715246
// MI455X (gfx1250) — compile-verified
//
#include <hip/hip_runtime.h>

// TRNN cell, MI455X (gfx1250, wave32).
// Bandwidth-bound: 3x 4096x4096 f32 GEMVs (~192 MiB @ 23.3 TB/s ~= 8.3 us floor).
// V_WMMA_F32_16X16X4_F32 performs the K-reduction; f32 kept end-to-end.

typedef __attribute__((ext_vector_type(2))) float v2f;
typedef __attribute__((ext_vector_type(8))) float v8f;

#define NN 4096

// One wave accumulates a 16-wide output tile of y = x . W over K in [kbeg, kbeg+kcount).
// A(16x4) = W^T tile  : lane m(0..15)/m+16 holds rows j0+m, VGPR0/1 = k-rows (kk, kk+1),
//                       lanes 16..31 carry kk = 2,3  (ISA 7.12.2, 32-bit 16x4 A layout).
// B(4x16)  = x[k+kk] broadcast across N -> every column of D equals the y tile.
// D layout : lane 0 holds y[j0..j0+7] in acc VGPRs 0..7, lane 16 holds y[j0+8..j0+15].
__device__ __forceinline__ v8f wmma_gemv_tile(const float* __restrict__ W,
                                              const float* __restrict__ x,
                                              int j0, int kbeg, int kcount) {
  const int lane = threadIdx.x & 31;
  const int m    = lane & 15;
  const int kk   = (lane >> 4) << 1;               // 0 (lanes 0-15) or 2 (lanes 16-31)
  const float* Wa = W + (size_t)(kbeg + kk) * NN + (j0 + m);
  const float* xb = x + kbeg + kk;
  v8f c0 = {0.f, 0.f, 0.f, 0.f, 0.f, 0.f, 0.f, 0.f};
  v8f c1 = c0;                                     // 2 chains hide XDL latency
  for (int i = 0; i < kcount; i += 8) {
    v2f a0, b0, a1, b1;
    a0[0] = Wa[0];
    a0[1] = Wa[NN];
    b0[0] = xb[0];
    b0[1] = xb[1];
    c0 = __builtin_amdgcn_wmma_f32_16x16x4_f32(false, a0, false, b0,
                                               (short)0, c0, false, false);
    a1[0] = Wa[4 * NN];
    a1[1] = Wa[5 * NN];
    b1[0] = xb[4];
    b1[1] = xb[5];
    c1 = __builtin_amdgcn_wmma_f32_16x16x4_f32(false, a1, false, b1,
                                               (short)0, c1, false, false);
    Wa += (size_t)8 * NN;
    xb += 8;
  }
  return c0 + c1;
}

// Dump this wave's 16 partial sums into LDS (lane 0 -> rows 0..7, lane 16 -> rows 8..15).
__device__ __forceinline__ void stash_partials(float* red, const v8f& c) {
  const int lane = threadIdx.x & 31;
  const int wave = threadIdx.x >> 5;
  if ((lane & 15) == 0) {
    const int base = wave * 16 + ((lane >> 4) << 3);
#pragma unroll
    for (int v = 0; v < 8; ++v) red[base + v] = c[v];
  }
}

// Kernel 1: h1 = tanh( t + x @ x_h_w + x_h_b ), t = collapsed TT term.
// 256 blocks x 512 threads; 16 waves split K into 16 chunks of 256.
__global__ __launch_bounds__(512) void trnn_hidden_kernel(
    const float* __restrict__ x, const float* __restrict__ core0,
    const float* __restrict__ core1, const float* __restrict__ x_h_w,
    const float* __restrict__ x_h_b, float* __restrict__ out) {
  __shared__ float red[16 * 16];
  const int tid  = threadIdx.x;
  const int wave = tid >> 5;
  const int j0   = blockIdx.x << 4;

  v8f c = wmma_gemv_tile(x_h_w, x, j0, wave * 256, 256);
  stash_partials(red, c);
  __syncthreads();

  if (tid < 16) {
    float s = 0.f;
#pragma unroll
    for (int w = 0; w < 16; ++w) s += red[w * 16 + tid];
    const int j = j0 + tid;
    // TT branch: h0 = e_4096 => t[j] = sum_r core0[65536+r] * core1[(65536+r)*4096 + j]
    float tt = 0.f;
#pragma unroll
    for (int r = 0; r < 16; ++r)
      tt = fmaf(core0[65536 + r], core1[(size_t)(65536 + r) * NN + j], tt);
    out[NN + j] = tanhf(s + tt + x_h_b[j]);
  }
}

// Kernel 2: z = sigmoid( h1 @ h_o_w + h_o_b + x @ x_o_w + x_o_b ).
// Waves 0-7: h1 GEMV (K chunks of 512); waves 8-15: x GEMV. Branch is wave-uniform
// so EXEC stays all-ones inside every WMMA.
__global__ __launch_bounds__(512) void trnn_out_kernel(
    const float* __restrict__ x, const float* __restrict__ h_o_w,
    const float* __restrict__ h_o_b, const float* __restrict__ x_o_w,
    const float* __restrict__ x_o_b, float* __restrict__ out) {
  __shared__ float red[16 * 16];
  const int tid  = threadIdx.x;
  const int wave = tid >> 5;
  const int j0   = blockIdx.x << 4;
  const float* h1 = out + NN;  // written by trnn_hidden_kernel

  v8f c;
  if (wave < 8)
    c = wmma_gemv_tile(h_o_w, h1, j0, wave * 512, 512);
  else
    c = wmma_gemv_tile(x_o_w, x, j0, (wave - 8) * 512, 512);
  stash_partials(red, c);
  __syncthreads();

  if (tid < 16) {
    float s = 0.f;
#pragma unroll
    for (int w = 0; w < 16; ++w) s += red[w * 16 + tid];
    const int j = j0 + tid;
    const float a = s + h_o_b[j] + x_o_b[j];
    out[j] = 1.f / (1.f + expf(-a));
  }
}

extern "C" void kernel_launch(void* const* d_in, const int* in_sizes, int n_in,
                              void* d_out, int out_size, void* d_ws, size_t ws_size,
                              hipStream_t stream) {
  (void)in_sizes; (void)n_in; (void)out_size; (void)d_ws; (void)ws_size;
  const float* x     = (const float*)d_in[0];
  const float* core0 = (const float*)d_in[1];
  const float* core1 = (const float*)d_in[2];
  const float* x_h_w = (const float*)d_in[3];
  const float* x_h_b = (const float*)d_in[4];
  const float* h_o_w = (const float*)d_in[5];
  const float* h_o_b = (const float*)d_in[6];
  const float* x_o_w = (const float*)d_in[7];
  const float* x_o_b = (const float*)d_in[8];
  float* out = (float*)d_out;

  dim3 grid(NN / 16);   // 256 blocks, one 16-column tile each
  dim3 block(512);      // 16 wave32s

  trnn_hidden_kernel<<<grid, block, 0, stream>>>(x, core0, core1, x_h_w, x_h_b, out);
  trnn_out_kernel<<<grid, block, 0, stream>>>(x, h_o_w, h_o_b, x_o_w, x_o_b, out);
}